// Poi2d_13572096655535
// MI455X (gfx1250) — compile-verified
//
#include <hip/hip_runtime.h>

// Elementwise: out[i] = -A * (x0*x0 - x0 + x1*x1 - x1)  with A = 1.0
//            =  A * ( x0*(1-x0) + x1*(1-x1) )
//
// Memory-bound streaming kernel (12 B / element, ~0.4 FLOP/B).
// Strategy: 128-bit NT loads/stores, 16 outputs per thread for MLP,
// 256-thread (8 x wave32) blocks. No WMMA: arithmetic intensity is ~1000x below
// the matrix-engine roofline; the win is pure HBM streaming.

typedef float float4v __attribute__((ext_vector_type(4)));

#define EPT 16  // output elements per thread (8x b128 loads, 4x b128 stores)

__global__ __launch_bounds__(256) void Poi2d_kernel(const float* __restrict__ in,
                                                    float* __restrict__ out,
                                                    int n) {
    const float A = 1.0f;
    const long long tid  = (long long)blockIdx.x * blockDim.x + threadIdx.x;
    const long long base = tid * EPT;

    if (base + EPT <= n) {
        // Fast path: fully vectorized, non-temporal 128-bit traffic.
        const float4v* inv  = (const float4v*)(in + 2 * base);  // 8 x float4 = 16 pairs
        float4v*       outv = (float4v*)(out + base);           // 4 x float4 = 16 outputs

        #pragma unroll
        for (int j = 0; j < 4; ++j) {
            // Each output float4 consumes two input float4s:
            // a = (x0[0], x1[0], x0[1], x1[1]),  b = (x0[2], x1[2], x0[3], x1[3])
            float4v a = __builtin_nontemporal_load(&inv[2 * j + 0]);
            float4v b = __builtin_nontemporal_load(&inv[2 * j + 1]);

            float4v r;
            r.x = A * __builtin_fmaf(a.x, 1.0f - a.x, a.y * (1.0f - a.y));
            r.y = A * __builtin_fmaf(a.z, 1.0f - a.z, a.w * (1.0f - a.w));
            r.z = A * __builtin_fmaf(b.x, 1.0f - b.x, b.y * (1.0f - b.y));
            r.w = A * __builtin_fmaf(b.z, 1.0f - b.z, b.w * (1.0f - b.w));

            __builtin_nontemporal_store(r, &outv[j]);
        }
    } else if (base < n) {
        // Scalar tail (never taken for N = 16777216, kept for safety).
        for (long long i = base; i < n; ++i) {
            float x0 = in[2 * i + 0];
            float x1 = in[2 * i + 1];
            out[i] = A * __builtin_fmaf(x0, 1.0f - x0, x1 * (1.0f - x1));
        }
    }
}

extern "C" void kernel_launch(void* const* d_in, const int* in_sizes, int n_in,
                              void* d_out, int out_size, void* d_ws, size_t ws_size,
                              hipStream_t stream) {
    (void)in_sizes; (void)n_in; (void)d_ws; (void)ws_size;

    const float* in  = (const float*)d_in[0];
    float*       out = (float*)d_out;
    const int n = out_size;  // number of output rows (= N)

    const int block = 256;                                   // 8 wave32s / block
    const long long threads = ((long long)n + EPT - 1) / EPT;
    const int grid = (int)((threads + block - 1) / block);

    Poi2d_kernel<<<grid, block, 0, stream>>>(in, out, n);
}